// CenterNet_6665789244043
// MI455X (gfx1250) — compile-verified
//
#include <hip/hip_runtime.h>

#define B_   16
#define C_   80
#define HH_  128
#define WW_  128
#define M_   32
#define DSF  4
#define IMGF 512.0f

#define NPIX (B_*HH_*WW_)        // 262144
#define OFF_VALUES 0
#define OFF_IDXS   (NPIX)
#define OFF_BBOX   (2*NPIX)
#define OFF_KEEP   (6*NPIX)
#define OFF_PFL    (7*NPIX)
#define OFF_SIZEL  (7*NPIX+1)
#define OFF_OFFL   (7*NPIX+2)
#define OFF_GTHM   (7*NPIX+3)

// workspace layout (floats)
#define WS_ROW  0      // 512: center cell row  (floor(cx/4) -- dim0 indexed by X per reference)
#define WS_COL  512    // 512: center cell col  (floor(cy/4))
#define WS_I2S  1024   // 512: 1/(2*sigma)
#define WS_SLS  1536   // 512: smooth-l1 size contribution (only if last occurrence of cell)
#define WS_SLO  2048   // 512: smooth-l1 offset contribution
#define WS_LAST 2560   // 512: 1.0 if this (b,m) is last writer of its cell (counts toward n)
#define WS_PART 3072   // 1280: per-plane focal partial sums

typedef __attribute__((ext_vector_type(2))) float v2f;
typedef __attribute__((ext_vector_type(8))) float v8f;

// ---- wave32 sum via V_WMMA_F32_16X16X4_F32: A holds lane values (K=0 / K=2 slots),
// B = ones => D[i][j] = rowsum(A_i); sum 8 acc VGPRs + fold lanes j <-> j+16.
__device__ __forceinline__ float wave_sum32(float v) {
  v2f a = {v, 0.0f};
  v2f b = {1.0f, 1.0f};
  v8f c = {0.f, 0.f, 0.f, 0.f, 0.f, 0.f, 0.f, 0.f};
  c = __builtin_amdgcn_wmma_f32_16x16x4_f32(false, a, false, b, (short)0, c, false, false);
  float s = c[0] + c[1] + c[2] + c[3] + c[4] + c[5] + c[6] + c[7];
  s += __shfl_xor(s, 16, 32);
  return s;  // total over 32 lanes, replicated
}

// ---- CDNA5 async memory->LDS copy (ASYNCcnt). Inline asm: portable across toolchains.
__device__ __forceinline__ void async_load_lds_b32(void* lds, const void* gaddr) {
  unsigned lo = (unsigned)(unsigned long long)lds;  // low 32 bits of generic ptr == LDS offset
  asm volatile("global_load_async_to_lds_b32 %0, %1, off"
               :
               : "v"(lo), "v"((unsigned long long)gaddr)
               : "memory");
}
__device__ __forceinline__ void wait_asynccnt0() {
  asm volatile("s_wait_asynccnt 0" ::: "memory");
}

__device__ __forceinline__ float smoothl1(float p, float g) {
  float d = fabsf(p - g);
  return (d < 1.0f) ? 0.5f * d * d : d - 0.5f;
}

// ===================== Kernel 1: peaks/values/idxs/keep + bbox decode =====================
__global__ __launch_bounds__(256) void k_detect(const float* __restrict__ hm,
                                                const float* __restrict__ sz,
                                                const float* __restrict__ osp,
                                                float* __restrict__ out) {
  __shared__ float tile[18 * 18];
  const int tx = threadIdx.x & 15, ty = threadIdx.x >> 4;
  const int bx = blockIdx.x, by = blockIdx.y, b = blockIdx.z;
  const int x = bx * 16 + tx, y = by * 16 + ty;
  const int pix = (b * HH_ + y) * WW_ + x;

  // ---- dense bbox decode (pts dim0 = row index, channel0 pairs with it) ----
  {
    const int i0 = ((b * 2 + 0) * HH_ + y) * WW_ + x;
    const int i1 = ((b * 2 + 1) * HH_ + y) * WW_ + x;
    float o0 = osp[i0], o1 = osp[i1];
    float s0 = sz[i0],  s1 = sz[i1];
    float c0 = (float)y * (float)DSF + o0;
    float c1 = (float)x * (float)DSF + o1;
    out[OFF_BBOX + pix * 4 + 0] = fminf(fmaxf(c0 - 0.5f * s0, 0.f), IMGF);
    out[OFF_BBOX + pix * 4 + 1] = fminf(fmaxf(c1 - 0.5f * s1, 0.f), IMGF);
    out[OFF_BBOX + pix * 4 + 2] = fminf(fmaxf(c0 + 0.5f * s0, 0.f), IMGF);
    out[OFF_BBOX + pix * 4 + 3] = fminf(fmaxf(c1 + 0.5f * s1, 0.f), IMGF);
  }

  // ---- 3x3 maxpool NMS, running max/argmax over classes ----
  float vbest = -__builtin_inff();
  int   cbest = 0;
  for (int c = 0; c < C_; ++c) {
    const float* plane = hm + (size_t)(b * C_ + c) * (HH_ * WW_);
    for (int e = threadIdx.x; e < 18 * 18; e += 256) {
      int gy = by * 16 - 1 + e / 18;
      int gx = bx * 16 - 1 + e % 18;
      if (gy >= 0 && gy < HH_ && gx >= 0 && gx < WW_) {
        async_load_lds_b32(&tile[e], &plane[gy * WW_ + gx]);  // async mem->LDS
      } else {
        tile[e] = -__builtin_inff();  // SAME padding identity for reduce_window(max)
      }
    }
    wait_asynccnt0();
    __syncthreads();
    float center = tile[(ty + 1) * 18 + (tx + 1)];
    float mx = -__builtin_inff();
    #pragma unroll
    for (int dy = 0; dy < 3; ++dy)
      #pragma unroll
      for (int dx = 0; dx < 3; ++dx)
        mx = fmaxf(mx, tile[(ty + dy) * 18 + (tx + dx)]);
    float p = (center == mx) ? center : 0.0f;     // peak or 0
    if (p > vbest) { vbest = p; cbest = c; }      // first-max wins (argmax semantics)
    __syncthreads();
  }
  out[OFF_VALUES + pix] = vbest;
  out[OFF_IDXS + pix]   = (float)cbest;
  out[OFF_KEEP + pix]   = (vbest > 0.0f) ? 1.0f : 0.0f;
}

// ===================== Kernel 2: per-(b,m) targets, dedup, center-pixel smooth-L1 ==========
__global__ __launch_bounds__(256) void k_targets(const float* __restrict__ gtb,
                                                 const float* __restrict__ sz,
                                                 const float* __restrict__ osp,
                                                 float* __restrict__ ws) {
  const int i = blockIdx.x * 256 + threadIdx.x;
  if (i >= B_ * M_) return;
  const int b = i / M_, m = i % M_;
  const float* g = gtb + (size_t)i * 4;
  float cx = 0.5f * (g[0] + g[2]);
  float cy = 0.5f * (g[1] + g[3]);
  float w = g[2] - g[0], h = g[3] - g[1];
  float r = floorf(cx * 0.25f);     // dim0 index comes from X center (reference convention)
  float c = floorf(cy * 0.25f);
  float offx = cx * 0.25f - r;
  float offy = cy * 0.25f - c;
  float sigma = fmaxf(w, h) * 0.25f;
  ws[WS_ROW + i] = r;
  ws[WS_COL + i] = c;
  ws[WS_I2S + i] = 1.0f / (2.0f * sigma);

  // last occurrence of this cell wins the scatter and is counted once in n
  const int ri = (int)r, ci = (int)c;
  bool last = true;
  for (int m2 = m + 1; m2 < M_; ++m2) {
    const float* g2 = gtb + ((size_t)b * M_ + m2) * 4;
    int r2 = (int)floorf(0.5f * (g2[0] + g2[2]) * 0.25f);
    int c2 = (int)floorf(0.5f * (g2[1] + g2[3]) * 0.25f);
    if (r2 == ri && c2 == ci) { last = false; break; }
  }
  float sls = 0.0f, slo = 0.0f;
  if (last) {
    int rc = min(max(ri, 0), HH_ - 1), cc = min(max(ci, 0), WW_ - 1);
    int i0 = ((b * 2 + 0) * HH_ + rc) * WW_ + cc;
    int i1 = ((b * 2 + 1) * HH_ + rc) * WW_ + cc;
    sls = smoothl1(sz[i0], w)    + smoothl1(sz[i1], h);
    slo = smoothl1(osp[i0], offx) + smoothl1(osp[i1], offy);
  }
  ws[WS_SLS + i]  = sls;
  ws[WS_SLO + i]  = slo;
  ws[WS_LAST + i] = last ? 1.0f : 0.0f;
}

// ===================== Kernel 3: gt_hm plane + focal-loss partials (per (b,c)) =============
__global__ __launch_bounds__(256) void k_gthm(const float* __restrict__ hm,
                                              const int* __restrict__ gtc,
                                              const float* __restrict__ ws,
                                              float* __restrict__ out,
                                              float* __restrict__ partials) {
  __shared__ int   cnt;
  __shared__ float s_r[M_], s_c[M_], s_i2s[M_];
  __shared__ float warp_part[8];
  const int plane = blockIdx.x;            // b*C + c
  const int b = plane / C_, cls = plane % C_;
  if (threadIdx.x == 0) {                  // serial build: deterministic list order
    int n = 0;
    for (int m = 0; m < M_; ++m) {
      if (gtc[b * M_ + m] == cls) {
        s_r[n]   = ws[WS_ROW + b * M_ + m];
        s_c[n]   = ws[WS_COL + b * M_ + m];
        s_i2s[n] = ws[WS_I2S + b * M_ + m];
        ++n;
      }
    }
    cnt = n;
  }
  __syncthreads();
  const int n = cnt;
  const float* hmp = hm + (size_t)plane * (HH_ * WW_);
  float* gp = out + OFF_GTHM + (size_t)plane * (HH_ * WW_);
  float acc = 0.0f;
  for (int px = threadIdx.x; px < HH_ * WW_; px += 256) {
    float fi = (float)(px >> 7);
    float fj = (float)(px & 127);
    float gmax = 0.0f;                     // segment_max(empty)->-inf then max(,0) => 0
    for (int k = 0; k < n; ++k) {
      float dr = fi - s_r[k], dc = fj - s_c[k];
      gmax = fmaxf(gmax, expf(-(dr * dr + dc * dc) * s_i2s[k]));
    }
    gp[px] = gmax;
    float hv = hmp[px];
    float t1 = 1.0f - gmax, t2 = t1 * t1, t4 = t2 * t2;
    float q2 = gmax * gmax, q4 = q2 * q2;
    float u = 1.0f - hv;
    acc += t4 * hv * hv * log1pf(-hv) + q4 * u * u * logf(hv);
  }
  float wsum = wave_sum32(acc);            // v_wmma_f32_16x16x4_f32 reduction
  const int lane = threadIdx.x & 31, wid = threadIdx.x >> 5;
  if (lane == 0) warp_part[wid] = wsum;
  __syncthreads();
  if (threadIdx.x == 0) {
    float t = 0.0f;
    for (int w = 0; w < 8; ++w) t += warp_part[w];  // fixed order: deterministic
    partials[plane] = t;
  }
}

// ===================== Kernel 4: final deterministic reduction -> 3 scalars ================
__global__ __launch_bounds__(256) void k_finalize(const float* __restrict__ ws,
                                                  float* __restrict__ out) {
  __shared__ float wp[8][4];
  float afl = 0.f, as = 0.f, ao = 0.f, an = 0.f;
  for (int i = threadIdx.x; i < 1280; i += 256) afl += ws[WS_PART + i];
  for (int i = threadIdx.x; i < 512; i += 256) {
    as += ws[WS_SLS + i];
    ao += ws[WS_SLO + i];
    an += ws[WS_LAST + i];
  }
  float rfl = wave_sum32(afl);
  float rs  = wave_sum32(as);
  float ro  = wave_sum32(ao);
  float rn  = wave_sum32(an);
  const int lane = threadIdx.x & 31, wid = threadIdx.x >> 5;
  if (lane == 0) { wp[wid][0] = rfl; wp[wid][1] = rs; wp[wid][2] = ro; wp[wid][3] = rn; }
  __syncthreads();
  if (threadIdx.x == 0) {
    float tf = 0.f, ts = 0.f, to = 0.f, tn = 0.f;
    for (int w = 0; w < 8; ++w) { tf += wp[w][0]; ts += wp[w][1]; to += wp[w][2]; tn += wp[w][3]; }
    out[OFF_PFL]   = -tf;
    out[OFF_SIZEL] = ts / (2.0f * tn) / tn;
    out[OFF_OFFL]  = to / (2.0f * tn) / tn;
  }
}

extern "C" void kernel_launch(void* const* d_in, const int* in_sizes, int n_in,
                              void* d_out, int out_size, void* d_ws, size_t ws_size,
                              hipStream_t stream) {
  const float* hm  = (const float*)d_in[0];
  const float* sz  = (const float*)d_in[1];
  const float* osp = (const float*)d_in[2];
  const float* gtb = (const float*)d_in[3];
  const int*   gtc = (const int*)d_in[4];
  float* out = (float*)d_out;
  float* ws  = (float*)d_ws;

  k_detect  <<<dim3(8, 8, 16), 256, 0, stream>>>(hm, sz, osp, out);
  k_targets <<<2, 256, 0, stream>>>(gtb, sz, osp, ws);
  k_gthm    <<<B_ * C_, 256, 0, stream>>>(hm, gtc, ws, out, ws + WS_PART);
  k_finalize<<<1, 256, 0, stream>>>(ws, out);
}